// MandoFanbeamFpjLayer_26740466385678
// MI455X (gfx1250) — compile-verified
//
#include <hip/hip_runtime.h>
#include <cstdint>

// ---- geometry constants (match reference) ----
#define K_VIEWS 128
#define K_NDET  512
#define K_IMG   256
#define K_NS    364                       // ceil(2R/STEP)+1, R = 128*sqrt(2)
#define K_SID   750.0f
#define K_SDD   1250.0f
#define K_DETE  1.2f
#define K_T0    568.98066401624376f       // SID - R
#define K_BSTEP 0.04908738521234052f      // 2*pi/128 radians per view
#define K_LSTRIDE 257                     // padded LDS row stride (floats): 1028 B -> bank-conflict free
#define K_LDS_BYTES (K_IMG * K_LSTRIDE * 4)   // 263168 B < 320 KB WGP LDS

// One block: one batch image resident in LDS (padded, 257 KB), 2 views x 512 dets = 1024 rays.
__global__ __launch_bounds__(1024, 1)
void fanbeam_fpj_kernel(const float* __restrict__ img, float* __restrict__ out) {
    extern __shared__ float lds[];                 // 256 rows x 257 floats (col 256 = pad, never read)
    const int tid    = threadIdx.x;
    const int b      = blockIdx.x >> 6;            // 64 view-chunks per batch
    const int vchunk = blockIdx.x & 63;
    const float* src_img = img + b * (K_IMG * K_IMG);

    // ---- stage full 256x256 image into padded LDS via CDNA5 async global->LDS DMA ----
    {
        const uint32_t lds_base = (uint32_t)(uintptr_t)lds;   // byte offset in LDS space
        const uint64_t gbase    = (uint64_t)(uintptr_t)src_img;
        #pragma unroll
        for (int k = 0; k < 16; ++k) {
            const uint32_t elt   = (uint32_t)(k * 1024 + tid);    // float4 index, lane-contiguous
            const uint32_t row   = elt >> 6;                      // 64 float4s per image row
            const uint32_t c16   = (elt & 63u) * 16u;
            const uint32_t laddr = lds_base + row * (K_LSTRIDE * 4u) + c16;
            const uint64_t gaddr = gbase + (uint64_t)elt * 16u;
            asm volatile("global_load_async_to_lds_b128 %0, %1, off"
                         :: "v"(laddr), "v"(gaddr) : "memory");
        }
        asm volatile("s_wait_asynccnt 0x0" ::: "memory");      // drain this wave's ASYNCcnt
    }
    __syncthreads();                                            // publish LDS to all waves

    // ---- per-ray geometry (registers only) ----
    const int v = vchunk * 2 + (tid >> 9);
    const int n = tid & 511;

    const float beta = (float)v * K_BSTEP;
    float sb, cb;
    sincosf(beta, &sb, &cb);
    const float srcx = K_SID * cb;
    const float srcy = K_SID * sb;
    const float u    = ((float)n - 255.5f) * K_DETE;
    const float rx   = -K_SDD * cb - u * sb;                   // det - src (unnormalized)
    const float ry   = -K_SDD * sb + u * cb;
    const float invL = rsqrtf(K_SDD * K_SDD + u * u);
    const float dx   = rx * invL;                              // d(jx)/ds
    const float dy   = -(ry * invL);                           // d(iy)/ds  (iy = 127.5 - py)
    // first sample point in pixel-index space
    const float p0j  = fmaf(K_T0, rx * invL, srcx) + 127.5f;   // jx at s=0
    const float p0i  = 127.5f - fmaf(K_T0, ry * invL, srcy);   // iy at s=0

    // ---- clip the sample range to the bilinear support box (-1, 256)^2 ----
    const float gdx = (fabsf(dx) < 1e-20f) ? copysignf(1e-20f, dx) : dx;
    const float gdy = (fabsf(dy) < 1e-20f) ? copysignf(1e-20f, dy) : dy;
    const float ix_ = 1.0f / gdx;
    const float iy_ = 1.0f / gdy;
    const float ax  = (-1.0f - p0j) * ix_;
    const float bx  = (256.0f - p0j) * ix_;
    const float ay  = (-1.0f - p0i) * iy_;
    const float by  = (256.0f - p0i) * iy_;
    const float smin = fmaxf(fminf(ax, bx), fminf(ay, by));
    const float smax = fminf(fmaxf(ax, bx), fmaxf(ay, by));
    const int s0 = max(0, (int)ceilf(smin));
    const int s1 = min(K_NS - 1, (int)floorf(smax));

    float acc = 0.0f;
    for (int s = s0; s <= s1; ++s) {
        const float fs  = (float)s;
        const float jx  = fmaf(fs, dx, p0j);
        const float iy  = fmaf(fs, dy, p0i);
        const float j0f = floorf(jx);
        const float i0f = floorf(iy);
        const int   j0  = (int)j0f;
        const int   i0  = (int)i0f;
        const float fj  = jx - j0f;
        const float fi  = iy - i0f;

        if ((unsigned)i0 < 255u && (unsigned)j0 < 255u) {
            // fully interior: 4 LDS dword gathers (pairs combine to ds_load_2addr_b32)
            const int base = i0 * K_LSTRIDE + j0;
            const float v00 = lds[base];
            const float v01 = lds[base + 1];
            const float v10 = lds[base + K_LSTRIDE];
            const float v11 = lds[base + K_LSTRIDE + 1];
            const float top = fmaf(fj, v01 - v00, v00);
            const float bot = fmaf(fj, v11 - v10, v10);
            acc += fmaf(fi, bot - top, top);
        } else {
            // border: per-corner zero-mask (matches reference clamp+mask semantics)
            const float v00 = ((unsigned)i0 < 256u && (unsigned)j0 < 256u)
                                  ? lds[i0 * K_LSTRIDE + j0] : 0.0f;
            const float v01 = ((unsigned)i0 < 256u && (unsigned)(j0 + 1) < 256u)
                                  ? lds[i0 * K_LSTRIDE + j0 + 1] : 0.0f;
            const float v10 = ((unsigned)(i0 + 1) < 256u && (unsigned)j0 < 256u)
                                  ? lds[(i0 + 1) * K_LSTRIDE + j0] : 0.0f;
            const float v11 = ((unsigned)(i0 + 1) < 256u && (unsigned)(j0 + 1) < 256u)
                                  ? lds[(i0 + 1) * K_LSTRIDE + j0 + 1] : 0.0f;
            const float top = fmaf(fj, v01 - v00, v00);
            const float bot = fmaf(fj, v11 - v10, v10);
            acc += fmaf(fi, bot - top, top);
        }
    }

    // STEP == 1.0 so no scale needed
    out[((b * K_VIEWS + v) << 9) + n] = acc;
}

extern "C" void kernel_launch(void* const* d_in, const int* in_sizes, int n_in,
                              void* d_out, int out_size, void* d_ws, size_t ws_size,
                              hipStream_t stream) {
    (void)in_sizes; (void)n_in; (void)out_size; (void)d_ws; (void)ws_size;
    const float* x = (const float*)d_in[0];   // [2, 256, 256] float32
    float* out = (float*)d_out;               // [2, 128, 512] float32
    // 128 blocks (2 batches x 64 view-chunks), 1024 threads, 257 KB dynamic LDS (padded rows)
    fanbeam_fpj_kernel<<<dim3(128), dim3(1024), K_LDS_BYTES, stream>>>(x, out);
}